// SwinTransformerBlock_22247930594030
// MI455X (gfx1250) — compile-verified
//
#include <hip/hip_runtime.h>
#include <math.h>

typedef __attribute__((ext_vector_type(2))) float v2f;
typedef __attribute__((ext_vector_type(8))) float v8f;

#define WSZ   7
#define SHIFT 3
#define NHEAD 4
#define HD    32
#define CCH   128
#define HW    49          // H == W == 49
#define NWIN  49          // windows per image (7x7)
#define NTOK  49          // tokens per window (7x7)
#define BATCH 48
#define MTOT  (BATCH*NWIN*NTOK)   // 115248, divisible by 16
#define HID   512

// ---------------- wave helpers (wave32) ----------------
__device__ __forceinline__ float wredsum(float v) {
#pragma unroll
  for (int o = 16; o >= 1; o >>= 1) v += __shfl_xor(v, o, 32);
  return v;
}
__device__ __forceinline__ float wredmax(float v) {
#pragma unroll
  for (int o = 16; o >= 1; o >>= 1) v = fmaxf(v, __shfl_xor(v, o, 32));
  return v;
}

__device__ __forceinline__ v8f wmma4(v2f a, v2f b, v8f c) {
  // V_WMMA_F32_16X16X4_F32 : full-precision fp32 matrix op on CDNA5
  return __builtin_amdgcn_wmma_f32_16x16x4_f32(
      /*neg_a=*/false, a, /*neg_b=*/false, b,
      /*c_mod=*/(short)0, c, /*reuse_a=*/false, /*reuse_b=*/false);
}

// region id used by the shift-mask construction (H=W=49, WS=7, SS=3)
__device__ __forceinline__ int regid(int win, int n) {
  int wh = win / 7, wwi = win % 7, r = n / 7, c = n % 7;
  int hh = wh * 7 + r, xx = wwi * 7 + c;
  int hr = (hh < 42) ? 0 : ((hh < 46) ? 1 : 2);
  int xr = (xx < 42) ? 0 : ((xx < 46) ? 1 : 2);
  return hr * 3 + xr;
}

// ---------------- kernel 1: LN1 + cyclic shift + window partition ---------
// one wave per window-ordered token; lane covers 4 channels
__global__ __launch_bounds__(256) void ln1_window_kernel(
    const float* __restrict__ x, const float* __restrict__ w,
    const float* __restrict__ b, float* __restrict__ XW) {
  int wid = threadIdx.x >> 5, lane = threadIdx.x & 31;
  int t = blockIdx.x * 8 + wid;          // window-ordered token id
  int n = t % NTOK;
  int bw = t / NTOK;
  int wwin = bw % NWIN;
  int bb = bw / NWIN;
  int wh = wwin / WSZ, wwi = wwin % WSZ, r = n / WSZ, c = n % WSZ;
  int hh = (wh * WSZ + r + SHIFT) % HW;    // undo roll(-SS): src = (i+SS)%H
  int ww2 = (wwi * WSZ + c + SHIFT) % HW;
  const float* xp = x + ((size_t)bb * HW * HW + (size_t)hh * HW + ww2) * CCH + lane * 4;
  float4 v = *(const float4*)xp;
  float mean = wredsum(v.x + v.y + v.z + v.w) * (1.0f / 128.0f);
  float d0 = v.x - mean, d1 = v.y - mean, d2 = v.z - mean, d3 = v.w - mean;
  float var = wredsum(d0 * d0 + d1 * d1 + d2 * d2 + d3 * d3) * (1.0f / 128.0f);
  float inv = rsqrtf(var + 1e-5f);
  float4 wv = *(const float4*)(w + lane * 4);
  float4 bv = *(const float4*)(b + lane * 4);
  float4 o;
  o.x = d0 * inv * wv.x + bv.x;
  o.y = d1 * inv * wv.y + bv.y;
  o.z = d2 * inv * wv.z + bv.z;
  o.w = d3 * inv * wv.w + bv.w;
  *(float4*)(XW + (size_t)t * CCH + lane * 4) = o;
}

// ------- generic fp32-WMMA GEMM: C = A @ Bw^T + bias (+fusion) -----------
// Each wave computes a 16(M) x 64(N) strip: one A-frag feeds 4 WMMAs.
// Software-pipelined: next k-step's frags load while current WMMAs execute.
// A: [M,K] row-major.  Bw: [N,K] row-major (so B^T[k][n] = Bw[n*K+k]).
// Requires N % 64 == 0 and K % 4 == 0 (384/128/512, K=128/512 all qualify).
// fuse: 0 none, 1 exact GELU, 2 add Res[row*N+col]
__global__ __launch_bounds__(256) void gemm_wmma_f32(
    const float* __restrict__ A, const float* __restrict__ Bw,
    const float* __restrict__ bias, float* __restrict__ C,
    const float* __restrict__ Res, int M, int N, int K, int fuse) {
  int wave = threadIdx.x >> 5, lane = threadIdx.x & 31;
  int strips_n = N >> 6;
  int nStrips = (M >> 4) * strips_n;
  int strip = blockIdx.x * 8 + wave;
  if (strip >= nStrips) return;             // wave-uniform exit (EXEC all-1 for WMMA)
  int tm = strip / strips_n, tn = strip % strips_n;
  int l15 = lane & 15;
  int koff = (lane >> 4) << 1;              // lanes 0-15: K 0,1 ; lanes 16-31: K 2,3
  const float* ap  = A  + (size_t)((tm << 4) + l15) * K + koff;
  const float* bp0 = Bw + (size_t)((tn << 6) + l15) * K + koff;
  const float* bp1 = bp0 + (size_t)16 * K;
  const float* bp2 = bp0 + (size_t)32 * K;
  const float* bp3 = bp0 + (size_t)48 * K;
  v8f acc0 = {}, acc1 = {}, acc2 = {}, acc3 = {};
  // pipeline prologue: stage first k-step frags
  v2f a  = *(const v2f*)(ap);
  v2f b0 = *(const v2f*)(bp0);
  v2f b1 = *(const v2f*)(bp1);
  v2f b2 = *(const v2f*)(bp2);
  v2f b3 = *(const v2f*)(bp3);
  for (int k = 4; k < K; k += 4) {
    v2f an  = *(const v2f*)(ap + k);       // issue next-step loads first:
    v2f bn0 = *(const v2f*)(bp0 + k);      // they stay outstanding while the
    v2f bn1 = *(const v2f*)(bp1 + k);      // four WMMAs below execute
    v2f bn2 = *(const v2f*)(bp2 + k);
    v2f bn3 = *(const v2f*)(bp3 + k);
    acc0 = wmma4(a, b0, acc0);
    acc1 = wmma4(a, b1, acc1);
    acc2 = wmma4(a, b2, acc2);
    acc3 = wmma4(a, b3, acc3);
    a = an; b0 = bn0; b1 = bn1; b2 = bn2; b3 = bn3;
  }
  // pipeline epilogue: last k-step
  acc0 = wmma4(a, b0, acc0);
  acc1 = wmma4(a, b1, acc1);
  acc2 = wmma4(a, b2, acc2);
  acc3 = wmma4(a, b3, acc3);

  int rbase = (tm << 4) + ((lane >> 4) << 3);   // C layout: VGPR r -> row r (+8 hi half)
#define EPILOGUE(ACC, SUB)                                                   \
  {                                                                          \
    int col = (tn << 6) + (SUB << 4) + l15;                                  \
    float bv = bias[col];                                                    \
    _Pragma("unroll") for (int r = 0; r < 8; r++) {                          \
      int row = rbase + r;                                                   \
      float vv = ACC[r] + bv;                                                \
      if (fuse == 1) vv = 0.5f * vv * (1.0f + erff(vv * 0.70710678118654752f)); \
      else if (fuse == 2) vv += Res[(size_t)row * N + col];                  \
      C[(size_t)row * N + col] = vv;                                         \
    }                                                                        \
  }
  EPILOGUE(acc0, 0)
  EPILOGUE(acc1, 1)
  EPILOGUE(acc2, 2)
  EPILOGUE(acc3, 3)
#undef EPILOGUE
}

// ---------------- kernel 3: windowed attention, one wave per (window,head) -
__global__ __launch_bounds__(32) void attn_kernel(
    const float* __restrict__ QKV, const float* __restrict__ rpb,
    float* __restrict__ O) {
  __shared__ float sQ[64][32];
  __shared__ float sK[64][32];
  __shared__ float sV[64][32];
  __shared__ float sS[64][64];
  int lane = threadIdx.x;
  int h = blockIdx.x & 3;
  int bw = blockIdx.x >> 2;          // b*NWIN + w
  int w = bw % NWIN;
  const float scale = 0.17677669529663687f;   // 32^-0.5

  // stage Q,K,V (pad rows 49..63 with zeros); QKV row = [3, NHEAD, HD]
  for (int j = 0; j < 64; j++) {
    float q = 0.f, k = 0.f, v = 0.f;
    if (j < NTOK) {
      const float* base = QKV + ((size_t)(bw * NTOK + j)) * (3 * CCH) + h * HD + lane;
      q = base[0] * scale;
      k = base[CCH];
      v = base[2 * CCH];
    }
    sQ[j][lane] = q; sK[j][lane] = k; sV[j][lane] = v;
  }
  __syncthreads();

  int l15 = lane & 15;
  int koff = (lane >> 4) << 1;
  int rbase = (lane >> 4) << 3;

  // S = Q @ K^T   (64x64 padded, K-dim = 32)
  for (int mt = 0; mt < 4; mt++) {
    int m = (mt << 4) + l15;
    for (int nt = 0; nt < 4; nt++) {
      int n = (nt << 4) + l15;
      v8f acc = {};
#pragma unroll
      for (int k0 = 0; k0 < 32; k0 += 4) {
        v2f a; a.x = sQ[m][k0 + koff]; a.y = sQ[m][k0 + koff + 1];
        v2f b; b.x = sK[n][k0 + koff]; b.y = sK[n][k0 + koff + 1];
        acc = wmma4(a, b, acc);
      }
#pragma unroll
      for (int r = 0; r < 8; r++) sS[(mt << 4) + rbase + r][(nt << 4) + l15] = acc[r];
    }
  }
  __syncthreads();

  // softmax rows 0..48 with relative-position bias + (faithful) shift mask
  for (int m = 0; m < NTOK; m++) {
    int r1 = m / 7, c1 = m % 7;
    int j1 = lane;                     // 0..31, always < 49
    int rj = j1 / 7, cj = j1 % 7;
    int idx = (r1 - rj + 6) * 13 + (c1 - cj + 6);
    float v1 = sS[m][j1] + rpb[idx * NHEAD + h] +
               ((regid(w, j1) != regid(m, j1)) ? -100.0f : 0.0f);
    int j2 = lane + 32;                // 32..63
    float v2 = -1e30f;
    if (j2 < NTOK) {
      int rj2 = j2 / 7, cj2 = j2 % 7;
      int idx2 = (r1 - rj2 + 6) * 13 + (c1 - cj2 + 6);
      v2 = sS[m][j2] + rpb[idx2 * NHEAD + h] +
           ((regid(w, j2) != regid(m, j2)) ? -100.0f : 0.0f);
    }
    float mx = wredmax(fmaxf(v1, v2));
    float e1 = __expf(v1 - mx);
    float e2 = (j2 < NTOK) ? __expf(v2 - mx) : 0.0f;
    float inv = 1.0f / wredsum(e1 + e2);
    sS[m][j1] = e1 * inv;
    sS[m][j2] = e2 * inv;              // zeros for padded columns
  }
  for (int m = NTOK; m < 64; m++) { sS[m][lane] = 0.f; sS[m][lane + 32] = 0.f; }
  __syncthreads();

  // O = P @ V   (64x32, K-dim = 64 padded)
  for (int mt = 0; mt < 4; mt++) {
    int m = (mt << 4) + l15;
    for (int nt = 0; nt < 2; nt++) {
      int n = (nt << 4) + l15;
      v8f acc = {};
#pragma unroll
      for (int k0 = 0; k0 < 64; k0 += 4) {
        v2f a; a.x = sS[m][k0 + koff]; a.y = sS[m][k0 + koff + 1];
        v2f b; b.x = sV[k0 + koff][n]; b.y = sV[k0 + koff + 1][n];
        acc = wmma4(a, b, acc);
      }
#pragma unroll
      for (int r = 0; r < 8; r++) {
        int row = (mt << 4) + rbase + r;
        if (row < NTOK)
          O[((size_t)(bw * NTOK + row)) * CCH + h * HD + (nt << 4) + l15] = acc[r];
      }
    }
  }
}

// -------- kernel 5: window reverse + roll back + residual + LN2 ----------
__global__ __launch_bounds__(256) void resid_ln2_kernel(
    const float* __restrict__ x, const float* __restrict__ OUT1,
    const float* __restrict__ w, const float* __restrict__ b,
    float* __restrict__ X2, float* __restrict__ H2) {
  int wid = threadIdx.x >> 5, lane = threadIdx.x & 31;
  int t = blockIdx.x * 8 + wid;          // original-order token id
  int pos = t % (HW * HW);
  int bb = t / (HW * HW);
  int hc = pos / HW, wc = pos % HW;
  int hh = (hc - SHIFT + HW) % HW;       // reverse roll(+SS)
  int ww2 = (wc - SHIFT + HW) % HW;
  int wh = hh / WSZ, r = hh % WSZ, wwi = ww2 / WSZ, c = ww2 % WSZ;
  int ts = (bb * NWIN + wh * WSZ + wwi) * NTOK + (r * WSZ + c);
  float4 xv = *(const float4*)(x + (size_t)t * CCH + lane * 4);
  float4 ov = *(const float4*)(OUT1 + (size_t)ts * CCH + lane * 4);
  float4 s;
  s.x = xv.x + ov.x; s.y = xv.y + ov.y; s.z = xv.z + ov.z; s.w = xv.w + ov.w;
  *(float4*)(X2 + (size_t)t * CCH + lane * 4) = s;
  float mean = wredsum(s.x + s.y + s.z + s.w) * (1.0f / 128.0f);
  float d0 = s.x - mean, d1 = s.y - mean, d2 = s.z - mean, d3 = s.w - mean;
  float var = wredsum(d0 * d0 + d1 * d1 + d2 * d2 + d3 * d3) * (1.0f / 128.0f);
  float inv = rsqrtf(var + 1e-5f);
  float4 wv = *(const float4*)(w + lane * 4);
  float4 bv = *(const float4*)(b + lane * 4);
  float4 o;
  o.x = d0 * inv * wv.x + bv.x;
  o.y = d1 * inv * wv.y + bv.y;
  o.z = d2 * inv * wv.z + bv.z;
  o.w = d3 * inv * wv.w + bv.w;
  *(float4*)(H2 + (size_t)t * CCH + lane * 4) = o;
}

// -------------------------------------------------------------------------
extern "C" void kernel_launch(void* const* d_in, const int* in_sizes, int n_in,
                              void* d_out, int out_size, void* d_ws, size_t ws_size,
                              hipStream_t stream) {
  const float* x       = (const float*)d_in[0];
  const float* norm1_w = (const float*)d_in[1];
  const float* norm1_b = (const float*)d_in[2];
  const float* qkv_w   = (const float*)d_in[3];
  const float* qkv_b   = (const float*)d_in[4];
  const float* proj_w  = (const float*)d_in[5];
  const float* proj_b  = (const float*)d_in[6];
  const float* rpb     = (const float*)d_in[7];
  const float* norm2_w = (const float*)d_in[8];
  const float* norm2_b = (const float*)d_in[9];
  const float* fc1_w   = (const float*)d_in[10];
  const float* fc1_b   = (const float*)d_in[11];
  const float* fc2_w   = (const float*)d_in[12];
  const float* fc2_b   = (const float*)d_in[13];
  float* out = (float*)d_out;

  // workspace layout (floats), with lifetime-based reuse:
  //   region0: XW (LN1 windowed)  ->  O (attn out)  ->  H2 (LN2 out)
  //   region1: QKV  ->  OUT1 (proj out)  ->  G (fc1 out, 512 cols)
  //   region2: X2 (residual-1 sum, original order)
  float* ws = (float*)d_ws;
  const size_t R0 = (size_t)MTOT * CCH;      // 14,751,744 floats
  const size_t R1 = (size_t)MTOT * HID;      // 59,006,976 floats (>= MTOT*384)
  float* XW   = ws;                          // region0
  float* O    = ws;                          // region0 (XW dead after QKV gemm)
  float* H2   = ws;                          // region0 (O dead after proj gemm)
  float* QKV  = ws + R0;                     // region1
  float* OUT1 = ws + R0;                     // region1 (QKV dead after attn)
  float* G    = ws + R0;                     // region1 (OUT1 dead after epilogue)
  float* X2   = ws + R0 + R1;                // region2

  const int WPB = 8;  // waves (16x64 strips) per block

  // 1) LN1 + shift + window partition
  ln1_window_kernel<<<MTOT / 8, 256, 0, stream>>>(x, norm1_w, norm1_b, XW);

  // 2) QKV = XW @ qkv_w^T + qkv_b   [115248 x 384]
  {
    int strips = (MTOT / 16) * (384 / 64);
    gemm_wmma_f32<<<(strips + WPB - 1) / WPB, 256, 0, stream>>>(
        XW, qkv_w, qkv_b, QKV, nullptr, MTOT, 384, CCH, 0);
  }

  // 3) windowed attention (one wave per window-head)
  attn_kernel<<<BATCH * NWIN * NHEAD, 32, 0, stream>>>(QKV, rpb, O);

  // 4) OUT1 = O @ proj_w^T + proj_b   [115248 x 128]
  {
    int strips = (MTOT / 16) * (CCH / 64);
    gemm_wmma_f32<<<(strips + WPB - 1) / WPB, 256, 0, stream>>>(
        O, proj_w, proj_b, OUT1, nullptr, MTOT, CCH, CCH, 0);
  }

  // 5) window reverse + roll + residual -> X2 ; LN2 -> H2
  resid_ln2_kernel<<<MTOT / 8, 256, 0, stream>>>(x, OUT1, norm2_w, norm2_b, X2, H2);

  // 6) G = GELU(H2 @ fc1_w^T + fc1_b)   [115248 x 512]
  {
    int strips = (MTOT / 16) * (HID / 64);
    gemm_wmma_f32<<<(strips + WPB - 1) / WPB, 256, 0, stream>>>(
        H2, fc1_w, fc1_b, G, nullptr, MTOT, HID, CCH, 1);
  }

  // 7) out = X2 + G @ fc2_w^T + fc2_b   [115248 x 128]
  {
    int strips = (MTOT / 16) * (CCH / 64);
    gemm_wmma_f32<<<(strips + WPB - 1) / WPB, 256, 0, stream>>>(
        G, fc2_w, fc2_b, out, X2, MTOT, CCH, HID, 2);
  }
}